// post_process_44530220925472
// MI455X (gfx1250) — compile-verified
//
#include <hip/hip_runtime.h>

#define BATCH 16
#define NN 512

typedef __attribute__((ext_vector_type(2))) float v2f;
typedef __attribute__((ext_vector_type(8))) float v8f;

// ---------------------------------------------------------------------------
// K1: vals = where(sigmoid(x)*mask > 0.5, sigmoid(x)*mask, 0); cost = -vals
// Pure bandwidth: 128-bit loads/stores, ~2us at 23.3 TB/s.
// ---------------------------------------------------------------------------
__global__ void k_fuse_cost(const float* __restrict__ x,
                            const float* __restrict__ mask,
                            float* __restrict__ cost) {
  int i = (blockIdx.x * blockDim.x + threadIdx.x) * 4;
  float4 xv = *(const float4*)(x + i);
  float4 mv = *(const float4*)(mask + i);
  float4 o;
  float s;
  s = mv.x / (1.0f + __expf(-xv.x)); o.x = (s > 0.5f) ? -s : 0.0f;
  s = mv.y / (1.0f + __expf(-xv.y)); o.y = (s > 0.5f) ? -s : 0.0f;
  s = mv.z / (1.0f + __expf(-xv.z)); o.z = (s > 0.5f) ? -s : 0.0f;
  s = mv.w / (1.0f + __expf(-xv.w)); o.w = (s > 0.5f) ? -s : 0.0f;
  *(float4*)(cost + i) = o;
}

// ---------------------------------------------------------------------------
// K2: Jonker-Volgenant LAP, one block (16 wave32) per batch.
// Column-private state (minv, v, used) lives in registers; only u/p/way in
// LDS. One __syncthreads per augmenting step (parity ping-pong argmin
// partials, redundant all-thread combine). The next step's cost row and
// u[i0] are prefetched right after the argmin resolves (provably not
// written this step). The first row of each outer iteration is staged via
// the CDNA5 async global->LDS engine, overlapped with t0's augment walk.
// ---------------------------------------------------------------------------
__global__ void __launch_bounds__(NN) k_lap(const float* __restrict__ cost,
                                            const float* __restrict__ mask,
                                            float* __restrict__ sec0) {
  const int b = blockIdx.x;
  const int t = threadIdx.x;       // 0..511
  const int j = t + 1;             // owned column 1..512
  const int wid = t >> 5;
  const float* C = cost + (size_t)b * NN * NN;

  __shared__ float u[NN + 1];
  __shared__ int   p[NN + 1];
  __shared__ int   way[NN + 1];
  __shared__ float rowbuf[NN];
  __shared__ float pv[2][16];
  __shared__ int   pj[2][16];

  float v_reg = 0.0f;              // dual v[j], column-private
  for (int e = t; e <= NN; e += NN) { u[e] = 0.0f; p[e] = 0; }

  // async prefetch of row 0 (first row read of outer iteration i=1)
  {
    unsigned lds_addr = (unsigned)(size_t)&rowbuf[t];
    unsigned long long ga = (unsigned long long)(C + t);
    asm volatile("global_load_async_to_lds_b32 %0, %1, off"
                 :: "v"(lds_addr), "v"(ga) : "memory");
  }
  int par = 0;
  __syncthreads();

  for (int i = 1; i <= NN; ++i) {
    float minv_reg = __builtin_inff();
    int   used_reg = 0;
    if (t == 0) p[0] = i;
    const int pj_reg = p[j];       // matched row of owned column (loop-stable)

    asm volatile("s_wait_asynccnt 0" ::: "memory");
    float c  = rowbuf[t];          // own element only -> no barrier needed
    float u0 = u[i];               // u[i0] for first step (i0 == i)

    int j0 = 0;
    for (;;) {
      if (j == j0) used_reg = 1;
      float m;
      if (!used_reg) {
        float cur = c - u0 - v_reg;
        if (cur < minv_reg) { minv_reg = cur; way[j] = j0; }
        m = minv_reg;
      } else {
        m = __builtin_inff();
      }

      // wave32 argmin (ties -> lowest column)
      float bv = m; int bj = j;
#pragma unroll
      for (int off = 16; off > 0; off >>= 1) {
        float ov = __shfl_down(bv, off, 32);
        int   oj = __shfl_down(bj, off, 32);
        if (ov < bv || (ov == bv && oj < bj)) { bv = ov; bj = oj; }
      }
      if ((t & 31) == 0) { pv[par][wid] = bv; pj[par][wid] = bj; }
      __syncthreads();             // the only barrier in the step

      // redundant all-thread combine (uniform result)
      float delta = pv[par][0]; int j1 = pj[par][0];
#pragma unroll
      for (int w = 1; w < 16; ++w) {
        float wv_ = pv[par][w]; int wj_ = pj[par][w];
        if (wv_ < delta || (wv_ == delta && wj_ < j1)) { delta = wv_; j1 = wj_; }
      }
      par ^= 1;

      // prefetch next step's row + u entry (not written this step)
      const int  i0n  = p[j1];
      const bool cont = (i0n != 0);
      float cn = 0.0f, un = 0.0f;
      if (cont) {
        cn = C[(size_t)(i0n - 1) * NN + t];
        un = u[i0n];
      }

      // dual updates
      if (used_reg) { u[pj_reg] += delta; v_reg -= delta; }
      else          { minv_reg  -= delta; }
      if (t == 0)   u[i] += delta;        // u[p[0]]

      j0 = j1;
      if (!cont) break;            // free column reached (uniform)
      c = cn; u0 = un;
    }

    // async prefetch of next outer iteration's first row (row index i),
    // overlapped with the serial augment walk below
    if (i < NN) {
      unsigned lds_addr = (unsigned)(size_t)&rowbuf[t];
      unsigned long long ga = (unsigned long long)(C + (size_t)i * NN + t);
      asm volatile("global_load_async_to_lds_b32 %0, %1, off"
                   :: "v"(lds_addr), "v"(ga) : "memory");
    }
    if (t == 0) {                  // augment along 'way'
      int jj = j0;
      while (jj != 0) { int jn = way[jj]; p[jj] = p[jn]; jj = jn; }
    }
    __syncthreads();               // way[] reuse + p[] visibility
  }

  // sec0[r][c] = (r == p[c+1]-1) ? mask[r][c] : 0
  const float* Mk = mask + (size_t)b * NN * NN;
  float* S = sec0 + (size_t)b * NN * NN;
  for (int e = t; e < NN * NN; e += NN) {
    int r = e >> 9, c = e & (NN - 1);
    S[e] = ((p[c + 1] - 1) == r) ? Mk[e] : 0.0f;
  }
}

// ---------------------------------------------------------------------------
// K3: out = sec0 + sec0^T per batch, one 16x16 tile per wave.
// T1 loaded in C-layout; T2 loaded (coalesced) into A-layout so the A
// math-matrix equals T2^T; 4 chained V_WMMA_F32_16X16X4_F32 with
// shifted-identity B slices accumulate D = T2^T + T1 -- the matrix pipe
// performs the cross-lane transpose redistribution and the add.
// ---------------------------------------------------------------------------
__global__ void __launch_bounds__(256) k_symm(const float* __restrict__ sec0,
                                              float* __restrict__ out) {
  const int wave = (blockIdx.x * blockDim.x + threadIdx.x) >> 5;
  const int lane = threadIdx.x & 31;
  const int b  = wave >> 10;           // 32*32 tiles per batch
  const int ti = (wave >> 5) & 31;
  const int tj = wave & 31;
  const float* S = sec0 + (size_t)b * NN * NN;
  float* O = out + (size_t)b * NN * NN;
  const int h = lane >> 4, n = lane & 15;

  v8f acc;
#pragma unroll
  for (int r = 0; r < 8; ++r)
    acc[r] = S[(size_t)(ti * 16 + r + 8 * h) * NN + tj * 16 + n];

#pragma unroll
  for (int c = 0; c < 4; ++c) {
    v2f a;
    a.x = S[(size_t)(tj * 16 + 4 * c + 2 * h    ) * NN + ti * 16 + n];
    a.y = S[(size_t)(tj * 16 + 4 * c + 2 * h + 1) * NN + ti * 16 + n];
    v2f bI;
    bI.x = (n == (4 * c + 2 * h    )) ? 1.0f : 0.0f;
    bI.y = (n == (4 * c + 2 * h + 1)) ? 1.0f : 0.0f;
    acc = __builtin_amdgcn_wmma_f32_16x16x4_f32(
        false, a, false, bI, (short)0, acc, false, false);
  }

#pragma unroll
  for (int r = 0; r < 8; ++r)
    O[(size_t)(ti * 16 + r + 8 * h) * NN + tj * 16 + n] = acc[r];
}

// ---------------------------------------------------------------------------
// K4: sequential row/col conflict resolution (matches lax.scan), one block
// per batch. Redundant all-thread combine + parity ping-pong partials:
// 3 barriers per scan step. Row-write -> col-read hazard is same-thread
// only (element (i,i)), so no barrier between the two phases.
// ---------------------------------------------------------------------------
__global__ void __launch_bounds__(NN) k_resolve(float* __restrict__ out) {
  const int b = blockIdx.x, t = threadIdx.x;
  const int wid = t >> 5;
  float* M = out + (size_t)b * NN * NN;
  __shared__ float psum[2][16], pmax[2][16];
  __shared__ int   parg[2][16];
  int par = 0;

  for (int i = 0; i < NN; ++i) {
    // ---- row i ----
    float rv = M[(size_t)i * NN + t];
    {
      float sv = rv, bv = rv; int bi = t;
#pragma unroll
      for (int off = 16; off > 0; off >>= 1) {
        sv += __shfl_down(sv, off, 32);
        float ov = __shfl_down(bv, off, 32);
        int   oi = __shfl_down(bi, off, 32);
        if (ov > bv || (ov == bv && oi < bi)) { bv = ov; bi = oi; }
      }
      if ((t & 31) == 0) { psum[par][wid] = sv; pmax[par][wid] = bv; parg[par][wid] = bi; }
    }
    __syncthreads();
    {
      float s = 0.0f, mb = pmax[par][0]; int ma = parg[par][0];
#pragma unroll
      for (int w = 0; w < 16; ++w) {
        s += psum[par][w];
        if (pmax[par][w] > mb || (pmax[par][w] == mb && parg[par][w] < ma)) {
          mb = pmax[par][w]; ma = parg[par][w];
        }
      }
      par ^= 1;
      if (s > 1.0f) M[(size_t)i * NN + t] = (t == ma) ? 1.0f : 0.0f;
    }

    // ---- col i (after row update; only (i,i) overlaps -> same thread) ----
    float cv = M[(size_t)t * NN + i];
    {
      float sv = cv, bv = cv; int bi = t;
#pragma unroll
      for (int off = 16; off > 0; off >>= 1) {
        sv += __shfl_down(sv, off, 32);
        float ov = __shfl_down(bv, off, 32);
        int   oi = __shfl_down(bi, off, 32);
        if (ov > bv || (ov == bv && oi < bi)) { bv = ov; bi = oi; }
      }
      if ((t & 31) == 0) { psum[par][wid] = sv; pmax[par][wid] = bv; parg[par][wid] = bi; }
    }
    __syncthreads();
    {
      float s = 0.0f, mb = pmax[par][0]; int ma = parg[par][0];
#pragma unroll
      for (int w = 0; w < 16; ++w) {
        s += psum[par][w];
        if (pmax[par][w] > mb || (pmax[par][w] == mb && parg[par][w] < ma)) {
          mb = pmax[par][w]; ma = parg[par][w];
        }
      }
      par ^= 1;
      if (s > 1.0f) M[(size_t)t * NN + i] = (t == ma) ? 1.0f : 0.0f;
    }
    __syncthreads();               // col writes visible before next row read
  }
}

// ---------------------------------------------------------------------------
extern "C" void kernel_launch(void* const* d_in, const int* in_sizes, int n_in,
                              void* d_out, int out_size, void* d_ws,
                              size_t ws_size, hipStream_t stream) {
  const float* x    = (const float*)d_in[0];
  const float* mask = (const float*)d_in[1];
  float* out  = (float*)d_out;
  float* cost = (float*)d_ws;                          // 16.8 MB
  float* sec0 = cost + (size_t)BATCH * NN * NN;        // 16.8 MB

  const int n = BATCH * NN * NN;
  k_fuse_cost<<<n / 4 / 256, 256, 0, stream>>>(x, mask, cost);
  k_lap<<<BATCH, NN, 0, stream>>>(cost, mask, sec0);
  k_symm<<<(BATCH * 32 * 32) / 8, 256, 0, stream>>>(sec0, out);
  k_resolve<<<BATCH, NN, 0, stream>>>(out);
}